// FactorizedVectorQuantize_19602230739352
// MI455X (gfx1250) — compile-verified
//
#include <hip/hip_runtime.h>
#include <stdint.h>

// ---------------- problem dims ----------------
#define B_    8
#define DIN   1024
#define T_    2048
#define K_    8192
#define DCB   8
#define BT    (B_ * T_)            // 16384
#define NCH   2                    // i-split for in_project MLP parallelism
#define COMMIT_W 0.005f
#define CBW      1.0f
#define EPS_     1e-12f

// ---------------- d_out layout (floats, reference return order) ----------
#define O_OUT 0                         // z_q_out [B, DIN, T]
#define O_CL  (B_ * DIN * T_)           // commit_loss [B]
#define O_QL  (O_CL + B_)               // codebook_loss [B]
#define O_IDX (O_QL + B_)               // indices [B, T] (as float values)
#define O_ZE  (O_IDX + BT)              // z_e [B, DCB, T]

// ---------------- workspace layout (floats) ------------------------------
#define W_WIN  0                        // w_in  [DCB][DIN]
#define W_WOUT (W_WIN + DCB * DIN)      // w_out [DIN][DCB]
#define W_CBN  (W_WOUT + DIN * DCB)     // normalized codebook [K][DCB]
#define W_ENC  (W_CBN + K_ * DCB)       // normalized enc [BT][DCB]
#define W_PART (W_ENC + BT * DCB)       // z_e partials [NCH][BT][DCB]
// total = 212992 + 262144 floats (~1.9 MB)

typedef __attribute__((ext_vector_type(2))) float v2f;
typedef __attribute__((ext_vector_type(8))) float v8f;

// =========================================================================
// Kernel 1: weight-norm prep + codebook normalize + zero loss accumulators
// =========================================================================
__global__ void vq_prep_kernel(const float* __restrict__ in_v,
                               const float* __restrict__ in_g,
                               const float* __restrict__ out_v,
                               const float* __restrict__ out_g,
                               const float* __restrict__ codebook,
                               float* __restrict__ w_in,
                               float* __restrict__ w_out,
                               float* __restrict__ cbn,
                               float* __restrict__ loss_slots /* 16 floats */) {
  __shared__ float red[256];
  const int tid = threadIdx.x;
  if (blockIdx.x == 0) {
    if (tid < 2 * B_) loss_slots[tid] = 0.0f;
    // w_in: 8 rows, each needs a 1024-wide sum of squares
    for (int o = 0; o < DCB; ++o) {
      float p = 0.f;
      for (int i = tid; i < DIN; i += 256) {
        float v = in_v[o * DIN + i];
        p += v * v;
      }
      red[tid] = p;
      __syncthreads();
      for (int s = 128; s > 0; s >>= 1) {
        if (tid < s) red[tid] += red[tid + s];
        __syncthreads();
      }
      float scale = in_g[o] / sqrtf(red[0]);
      __syncthreads();  // everyone read red[0] before next-iter overwrite
      for (int i = tid; i < DIN; i += 256)
        w_in[o * DIN + i] = in_v[o * DIN + i] * scale;
    }
  } else {
    // one thread per row: 1024 w_out rows then 8192 codebook rows (8 wide)
    int row = (blockIdx.x - 1) * 256 + tid;
    if (row < DIN) {
      const float* v = out_v + (size_t)row * DCB;
      float s = 0.f;
#pragma unroll
      for (int d = 0; d < DCB; ++d) s += v[d] * v[d];
      float sc = out_g[row] / sqrtf(s);
#pragma unroll
      for (int d = 0; d < DCB; ++d) w_out[row * DCB + d] = v[d] * sc;
    } else if (row < DIN + K_) {
      int r = row - DIN;
      const float* v = codebook + (size_t)r * DCB;
      float s = 0.f;
#pragma unroll
      for (int d = 0; d < DCB; ++d) s += v[d] * v[d];
      float sc = 1.0f / fmaxf(sqrtf(s), EPS_);
#pragma unroll
      for (int d = 0; d < DCB; ++d) cbn[r * DCB + d] = v[d] * sc;
    }
  }
}

// =========================================================================
// Kernel 2a: in_project partial sums. The 67 MB z read is the roofline
// floor; split the 1024-deep channel reduction into NCH chunks so enough
// loads are in flight (waves x LOADcnt x 128B) to cover HBM latency.
// =========================================================================
__global__ void vq_inproj_partial(const float* __restrict__ z,
                                  const float* __restrict__ w_in,
                                  float* __restrict__ part) {  // [NCH][BT][DCB]
  __shared__ __align__(16) float sw[(DIN / NCH) * DCB];  // w_in slice, [i][o]
  const int tid = threadIdx.x;  // 128
  const int b = blockIdx.y;
  const int c = blockIdx.z;
  const int t = (blockIdx.x << 7) + tid;
  const int ibase = c * (DIN / NCH);

  for (int x = tid; x < (DIN / NCH) * DCB; x += 128) {
    int o = x & 7, i = x >> 3;
    sw[x] = w_in[o * DIN + ibase + i];
  }
  __syncthreads();

  float4 a0 = {0.f, 0.f, 0.f, 0.f}, a1 = {0.f, 0.f, 0.f, 0.f};
  const float* zp = z + ((size_t)b * DIN + ibase) * T_ + t;
  for (int i = 0; i < DIN / NCH; ++i) {
    float zv = zp[(size_t)i * T_];
    if ((i & 15) == 0 && i + 16 < DIN / NCH)
      __builtin_prefetch(zp + (size_t)(i + 16) * T_, 0, 0);  // global_prefetch_b8
    const float4 w0 = *(const float4*)&sw[i * 8];
    const float4 w1 = *(const float4*)&sw[i * 8 + 4];
    a0.x += w0.x * zv; a0.y += w0.y * zv; a0.z += w0.z * zv; a0.w += w0.w * zv;
    a1.x += w1.x * zv; a1.y += w1.y * zv; a1.z += w1.z * zv; a1.w += w1.w * zv;
  }
  float* pp = part + ((size_t)c * BT + (size_t)b * T_ + t) * DCB;
  *(float4*)(pp) = a0;
  *(float4*)(pp + 4) = a1;
}

// =========================================================================
// Kernel 2b: combine partials -> z_e (+bias) and L2-normalized enc.
// =========================================================================
__global__ void vq_ze_combine(const float* __restrict__ part,
                              const float* __restrict__ in_b,
                              float* __restrict__ ze_out,  // [B][DCB][T]
                              float* __restrict__ enc) {   // [BT][DCB]
  const int tid = threadIdx.x;  // 128
  const int b = blockIdx.x >> 4;
  const int t = ((blockIdx.x & 15) << 7) + tid;
  const size_t n = (size_t)b * T_ + t;

  float ze[DCB];
#pragma unroll
  for (int o = 0; o < DCB; ++o) ze[o] = in_b[o];
#pragma unroll
  for (int c = 0; c < NCH; ++c) {
    const float4 p0 = *(const float4*)(part + ((size_t)c * BT + n) * DCB);
    const float4 p1 = *(const float4*)(part + ((size_t)c * BT + n) * DCB + 4);
    ze[0] += p0.x; ze[1] += p0.y; ze[2] += p0.z; ze[3] += p0.w;
    ze[4] += p1.x; ze[5] += p1.y; ze[6] += p1.z; ze[7] += p1.w;
  }
  float ss = 0.f;
#pragma unroll
  for (int o = 0; o < DCB; ++o) ss += ze[o] * ze[o];
  const float inv = 1.0f / fmaxf(sqrtf(ss), EPS_);

  const size_t zbase = (size_t)b * DCB * T_ + t;
#pragma unroll
  for (int o = 0; o < DCB; ++o) ze_out[zbase + (size_t)o * T_] = ze[o];

  float4 e0 = {ze[0] * inv, ze[1] * inv, ze[2] * inv, ze[3] * inv};
  float4 e1 = {ze[4] * inv, ze[5] * inv, ze[6] * inv, ze[7] * inv};
  float* ep = enc + n * DCB;
  *(float4*)(ep) = e0;
  *(float4*)(ep + 4) = e1;
}

// =========================================================================
// Kernel 3: nearest neighbor via V_WMMA_F32_16X16X4_F32.
// Unit vectors -> argmin dist == argmax dot. Block = 256 threads (8 waves);
// each wave owns 16 enc rows (fixed A operand). Normalized codebook is
// double-buffered through LDS with CDNA5 async copies: issue chunk c+1,
// s_wait_asynccnt 8 retires chunk c (async loads complete in order), then
// sweep 64 16-wide B tiles. Tile loop unrolled x2 -> two independent WMMA
// accumulation chains in flight.
// =========================================================================
__global__ void vq_nn_wmma_kernel(const float* __restrict__ enc,
                                  const float* __restrict__ cbn,
                                  float* __restrict__ out_idx) {
  __shared__ __align__(16) float scb[2][1024 * DCB];  // 2 x 32 KB ping-pong
  __shared__ float sval[256 * 8];
  __shared__ int   sidx[256 * 8];

  const int tid  = threadIdx.x;
  const int wave = tid >> 5;
  const int lane = tid & 31;
  const int nloc = lane & 15;
  const bool hi  = lane >= 16;
  const int rowbase = blockIdx.x * 128 + wave * 16;

  // A operand (16x4 f32, two K-halves): lanes0-15 hold K=0,1 / lanes16-31 K=2,3
  const float* er = enc + (size_t)(rowbase + nloc) * DCB;
  const int ka = hi ? 2 : 0;
  v2f a_lo = {er[ka], er[ka + 1]};          // K = 0..3
  v2f a_hi = {er[4 + ka], er[4 + ka + 1]};  // K = 4..7

  float best[8];
  int bidx[8];
#pragma unroll
  for (int r = 0; r < 8; ++r) { best[r] = -1e30f; bidx[r] = 0; }

  const int kb = hi ? 1 : 0;  // B 4x16: V0 = rows K0(K1), V1 = rows K2(K3)

  // issue chunk 0 (8 async b128 loads per thread)
  {
    const float* src = cbn;
    for (int i = tid; i < 2048; i += 256) {
      uint32_t laddr = (uint32_t)(uintptr_t)(&scb[0][i * 4]);
      uint64_t gaddr = (uint64_t)(uintptr_t)(src + i * 4);
      asm volatile("global_load_async_to_lds_b128 %0, %1, off"
                   :: "v"(laddr), "v"(gaddr) : "memory");
    }
  }

  for (int chunk = 0; chunk < K_ / 1024; ++chunk) {
    if (chunk < K_ / 1024 - 1) {
      // prefetch next chunk into the other buffer (was fully read by the
      // barrier that ended iteration chunk-1)
      const float* src = cbn + (size_t)(chunk + 1) * 1024 * DCB;
      float* dst = scb[(chunk + 1) & 1];
      for (int i = tid; i < 2048; i += 256) {
        uint32_t laddr = (uint32_t)(uintptr_t)(dst + i * 4);
        uint64_t gaddr = (uint64_t)(uintptr_t)(src + i * 4);
        asm volatile("global_load_async_to_lds_b128 %0, %1, off"
                     :: "v"(laddr), "v"(gaddr) : "memory");
      }
      // async loads retire in order: <=8 outstanding means chunk's batch done
      asm volatile("s_wait_asynccnt 8" ::: "memory");
    } else {
      asm volatile("s_wait_asynccnt 0" ::: "memory");
    }
    __syncthreads();

    const float* buf = scb[chunk & 1];
#pragma unroll 2
    for (int tile = 0; tile < 64; ++tile) {
      const float* cr = &buf[(tile * 16 + nloc) * DCB];
      v2f b_lo = {cr[kb], cr[kb + 2]};
      v2f b_hi = {cr[kb + 4], cr[kb + 6]};
      v8f c = {0.f, 0.f, 0.f, 0.f, 0.f, 0.f, 0.f, 0.f};
      c = __builtin_amdgcn_wmma_f32_16x16x4_f32(false, a_lo, false, b_lo,
                                                (short)0, c, false, false);
      c = __builtin_amdgcn_wmma_f32_16x16x4_f32(false, a_hi, false, b_hi,
                                                (short)0, c, false, false);
      const int cidx = chunk * 1024 + tile * 16 + nloc;
#pragma unroll
      for (int r = 0; r < 8; ++r)
        if (c[r] > best[r]) { best[r] = c[r]; bidx[r] = cidx; }  // strict: keep lowest idx
    }
    __syncthreads();
  }

  // cross-lane argmax: row m of a wave lives in lanes {0..15 | 16..31}, vgpr m&7
#pragma unroll
  for (int r = 0; r < 8; ++r) { sval[tid * 8 + r] = best[r]; sidx[tid * 8 + r] = bidx[r]; }
  __syncthreads();
  if (tid < 128) {
    int m = tid & 15;
    int w = tid >> 4;
    int l0 = w * 32 + (m < 8 ? 0 : 16);
    int r = m & 7;
    float bv = -1e30f;
    int bi = 0x7fffffff;
    for (int j = 0; j < 16; ++j) {
      float v = sval[(l0 + j) * 8 + r];
      int   i = sidx[(l0 + j) * 8 + r];
      if (v > bv || (v == bv && i < bi)) { bv = v; bi = i; }
    }
    out_idx[blockIdx.x * 128 + w * 16 + m] = (float)bi;
  }
}

// =========================================================================
// Kernel 4: commit/codebook losses + out_project (STE forward value = z_q).
// Writes the 67 MB output with coalesced lanes; w_out broadcast from LDS.
// =========================================================================
__global__ void vq_outproj_kernel(const float* __restrict__ codebook,
                                  const float* __restrict__ w_out,
                                  const float* __restrict__ out_b,
                                  const float* __restrict__ ze,     // [B][DCB][T]
                                  const float* __restrict__ idxf,   // [BT]
                                  float* __restrict__ out,          // [B][DIN][T]
                                  float* __restrict__ commit,
                                  float* __restrict__ cbl) {
  __shared__ __align__(16) float sw[DIN * DCB];  // 32 KB
  __shared__ float sb[DIN];
  __shared__ float red[128];
  const int tid = threadIdx.x;  // 128
  const int b = blockIdx.x >> 4;
  const int t = ((blockIdx.x & 15) << 7) + tid;

  for (int x = tid; x < DIN * DCB; x += 128) sw[x] = w_out[x];
  for (int x = tid; x < DIN; x += 128) sb[x] = out_b[x];
  __syncthreads();

  const int idx = (int)idxf[(size_t)b * T_ + t];
  const float4 q0 = *(const float4*)(codebook + (size_t)idx * DCB);
  const float4 q1 = *(const float4*)(codebook + (size_t)idx * DCB + 4);

  // losses: sum over d of (z_e - z_q)^2
  const float* zep = ze + (size_t)b * DCB * T_ + t;
  float part = 0.f, d;
  d = zep[0 * T_] - q0.x; part += d * d;
  d = zep[1 * T_] - q0.y; part += d * d;
  d = zep[2 * T_] - q0.z; part += d * d;
  d = zep[3 * T_] - q0.w; part += d * d;
  d = zep[4 * T_] - q1.x; part += d * d;
  d = zep[5 * T_] - q1.y; part += d * d;
  d = zep[6 * T_] - q1.z; part += d * d;
  d = zep[7 * T_] - q1.w; part += d * d;
  red[tid] = part;
  __syncthreads();
  for (int s = 64; s > 0; s >>= 1) {
    if (tid < s) red[tid] += red[tid + s];
    __syncthreads();
  }
  if (tid == 0) {
    const float inv_n = 1.0f / (float)(DCB * T_);
    atomicAdd(&commit[b], red[0] * (COMMIT_W * inv_n));
    atomicAdd(&cbl[b], red[0] * (CBW * inv_n));
  }

  // out_project: 1024 rows x dot8 per t
  float* op = out + (size_t)b * DIN * T_ + t;
  for (int o = 0; o < DIN; ++o) {
    const float4 w0 = *(const float4*)&sw[o * 8];
    const float4 w1 = *(const float4*)&sw[o * 8 + 4];
    float acc = sb[o];
    acc += w0.x * q0.x + w0.y * q0.y + w0.z * q0.z + w0.w * q0.w;
    acc += w1.x * q1.x + w1.y * q1.y + w1.z * q1.z + w1.w * q1.w;
    op[(size_t)o * T_] = acc;
  }
}

// =========================================================================
extern "C" void kernel_launch(void* const* d_in, const int* in_sizes, int n_in,
                              void* d_out, int out_size, void* d_ws, size_t ws_size,
                              hipStream_t stream) {
  const float* z        = (const float*)d_in[0];
  const float* in_v     = (const float*)d_in[1];
  const float* in_g     = (const float*)d_in[2];
  const float* in_b     = (const float*)d_in[3];
  const float* out_v    = (const float*)d_in[4];
  const float* out_g    = (const float*)d_in[5];
  const float* out_b    = (const float*)d_in[6];
  const float* codebook = (const float*)d_in[7];

  float* outf = (float*)d_out;
  float* wsf  = (float*)d_ws;

  float* w_in  = wsf + W_WIN;
  float* w_out = wsf + W_WOUT;
  float* cbn   = wsf + W_CBN;
  float* enc   = wsf + W_ENC;
  float* partb = wsf + W_PART;

  // 1) weight norms + codebook normalize + zero loss slots
  vq_prep_kernel<<<1 + (DIN + K_ + 255) / 256, 256, 0, stream>>>(
      in_v, in_g, out_v, out_g, codebook, w_in, w_out, cbn, outf + O_CL);

  // 2a) in_project partial sums (i-split for memory-level parallelism)
  vq_inproj_partial<<<dim3(T_ / 128, B_, NCH), 128, 0, stream>>>(z, w_in, partb);

  // 2b) combine -> z_e (to d_out) + normalized enc (to ws)
  vq_ze_combine<<<B_ * (T_ / 128), 128, 0, stream>>>(partb, in_b, outf + O_ZE, enc);

  // 3) WMMA nearest-neighbor argmax -> indices (as float)
  vq_nn_wmma_kernel<<<BT / 128, 256, 0, stream>>>(enc, cbn, outf + O_IDX);

  // 4) losses + out_project
  vq_outproj_kernel<<<B_ * (T_ / 128), 128, 0, stream>>>(
      codebook, w_out, out_b, outf + O_ZE, outf + O_IDX,
      outf + O_OUT, outf + O_CL, outf + O_QL);
}